// GAT_3212635537950
// MI455X (gfx1250) — compile-verified
//
#include <hip/hip_runtime.h>
#include <hip/hip_bf16.h>

typedef __attribute__((ext_vector_type(2))) float v2f;
typedef __attribute__((ext_vector_type(8))) float v8f;

#define IN_FT  512
#define OUT_FT 128
#define NEG_SLOPE 0.2f

// ---------------------------------------------------------------------------
// Init: zero the output accumulator, per-node max (ordered-uint -inf) & denom
// ---------------------------------------------------------------------------
__global__ void gat_init_kernel(float* __restrict__ out,
                                unsigned* __restrict__ maxb,
                                float* __restrict__ denom, int N) {
    int i = blockIdx.x * blockDim.x + threadIdx.x;
    if (i < N * OUT_FT) out[i] = 0.0f;
    if (i < N) { maxb[i] = 0u; denom[i] = 0.0f; }
}

// ---------------------------------------------------------------------------
// Fused GEMM1 + GEMM2 with fp32 WMMA (V_WMMA_F32_16X16X4_F32).
// One block = 8 wave32 waves = one 16-row slab of nodes, all 128 out features.
//   GEMM1: x = seq @ W_fc^T   (K = 512), tile staged in LDS
//   GEMM2: h = x  @ W_gat^T   (K = 128), A read from LDS
// A layout (16x4 fp32): lanes 0-15 -> M=lane, K={0,1}; lanes 16-31 -> K={2,3}
// B layout (4x16 fp32): VGPR0 rows K0/K2, VGPR1 rows K1/K3, N = lane%16
// C/D layout: VGPR v -> row (v + (lane<16?0:8)), col = lane%16
// ---------------------------------------------------------------------------
__global__ void gat_gemm_fused_kernel(const float* __restrict__ seq,
                                      const float* __restrict__ Wfc,
                                      const float* __restrict__ Wgat,
                                      float* __restrict__ h, int N) {
    __shared__ float xt[16][132];          // 16x128 tile, padded pitch (no bank conflicts)

    const int tileM = blockIdx.x;          // node slab
    const int tileN = threadIdx.x >> 5;    // 0..7 : which 16 output features
    const int lane  = threadIdx.x & 31;
    const int l16   = lane & 15;
    const int koff  = (lane < 16) ? 0 : 2; // K sub-offset per ISA A/B layout
    const int rsel  = (lane < 16) ? 0 : 8; // C/D row split

    // ---- GEMM1: 16x16 tile of x = seq @ Wfc^T, K = 512 ----
    const float* arow = seq + (size_t)(tileM * 16 + l16) * IN_FT;
    const float* bcol = Wfc + (size_t)(tileN * 16 + l16) * IN_FT;

    v8f c1 = {};
#pragma unroll 8
    for (int k = 0; k < IN_FT; k += 4) {
        v2f a = *(const v2f*)(arow + k + koff);
        v2f b = *(const v2f*)(bcol + k + koff);
        c1 = __builtin_amdgcn_wmma_f32_16x16x4_f32(false, a, false, b,
                                                   (short)0, c1, false, false);
    }
    // stage x tile into LDS (row = local node, col = feature)
#pragma unroll
    for (int v = 0; v < 8; ++v)
        xt[v + rsel][tileN * 16 + l16] = c1[v];
    __syncthreads();

    // ---- GEMM2: 16x16 tile of h = x @ Wgat^T, K = 128, A from LDS ----
    const float* bcol2 = Wgat + (size_t)(tileN * 16 + l16) * OUT_FT;
    v8f c2 = {};
#pragma unroll
    for (int k = 0; k < OUT_FT; k += 4) {
        v2f a;
        a.x = xt[l16][k + koff];
        a.y = xt[l16][k + koff + 1];
        v2f b = *(const v2f*)(bcol2 + k + koff);
        c2 = __builtin_amdgcn_wmma_f32_16x16x4_f32(false, a, false, b,
                                                   (short)0, c2, false, false);
    }
#pragma unroll
    for (int v = 0; v < 8; ++v)
        h[(size_t)(tileM * 16 + v + rsel) * OUT_FT + tileN * 16 + l16] = c2[v];
}

// ---------------------------------------------------------------------------
// Per-node attention logits: a_s[n] = h[n,:]·att_src, a_d[n] = h[n,:]·att_dst
// One wave32 per node, shuffle reduction.
// ---------------------------------------------------------------------------
__global__ void gat_att_kernel(const float* __restrict__ h,
                               const float* __restrict__ att_src,
                               const float* __restrict__ att_dst,
                               float* __restrict__ a_s, float* __restrict__ a_d,
                               int N) {
    int n    = blockIdx.x * (blockDim.x >> 5) + (threadIdx.x >> 5);
    int lane = threadIdx.x & 31;
    if (n >= N) return;
    const float* hr = h + (size_t)n * OUT_FT;
    float s = 0.0f, d = 0.0f;
#pragma unroll
    for (int j = lane; j < OUT_FT; j += 32) {
        float hv = hr[j];
        s += hv * att_src[j];
        d += hv * att_dst[j];
    }
#pragma unroll
    for (int off = 16; off > 0; off >>= 1) {
        s += __shfl_xor(s, off, 32);
        d += __shfl_xor(d, off, 32);
    }
    if (lane == 0) { a_s[n] = s; a_d[n] = d; }
}

__device__ __forceinline__ unsigned ord_enc(float f) {
    unsigned b = __float_as_uint(f);
    return (b & 0x80000000u) ? ~b : (b | 0x80000000u);
}
__device__ __forceinline__ float ord_dec(unsigned u) {
    unsigned b = (u & 0x80000000u) ? (u & 0x7FFFFFFFu) : ~u;
    return __uint_as_float(b);
}

// ---------------------------------------------------------------------------
// Edge pass 1: e = leaky_relu(a_s[src] + a_d[dst]); segment-max via atomicMax
// Edges [0,E) come from edge_index (int64), edges [E, E+N) are self loops.
// ---------------------------------------------------------------------------
__global__ void gat_edge_max_kernel(const long long* __restrict__ ei,
                                    const float* __restrict__ a_s,
                                    const float* __restrict__ a_d,
                                    float* __restrict__ ebuf,
                                    unsigned* __restrict__ maxb,
                                    int E, int EP) {
    int i = blockIdx.x * blockDim.x + threadIdx.x;
    if (i >= EP) return;
    long long s, d;
    if (i < E) { s = ei[i]; d = ei[(size_t)E + i]; }
    else       { s = d = i - E; }
    float e = a_s[s] + a_d[d];
    e = (e >= 0.0f) ? e : NEG_SLOPE * e;
    ebuf[i] = e;
    atomicMax(&maxb[d], ord_enc(e));
}

// ---------------------------------------------------------------------------
// Edge pass 2: p = exp(e - max[dst]); denom[dst] += p
// ---------------------------------------------------------------------------
__global__ void gat_edge_exp_kernel(const long long* __restrict__ ei,
                                    float* __restrict__ ebuf,
                                    const unsigned* __restrict__ maxb,
                                    float* __restrict__ denom,
                                    int E, int EP) {
    int i = blockIdx.x * blockDim.x + threadIdx.x;
    if (i >= EP) return;
    long long d = (i < E) ? ei[(size_t)E + i] : (long long)(i - E);
    float p = __expf(ebuf[i] - ord_dec(maxb[d]));
    ebuf[i] = p;
    atomicAdd(&denom[d], p);
}

// ---------------------------------------------------------------------------
// Edge pass 3: out[dst,:] += (p/denom[dst]) * h[src,:]
// One wave32 per edge; each lane handles a float4 of features.
// ---------------------------------------------------------------------------
__global__ void gat_edge_scatter_kernel(const long long* __restrict__ ei,
                                        const float* __restrict__ ebuf,
                                        const float* __restrict__ denom,
                                        const float* __restrict__ h,
                                        float* __restrict__ out,
                                        int E, int EP) {
    int w    = blockIdx.x * (blockDim.x >> 5) + (threadIdx.x >> 5);
    int lane = threadIdx.x & 31;
    if (w >= EP) return;
    long long s, d;
    if (w < E) { s = ei[w]; d = ei[(size_t)E + w]; }
    else       { s = d = w - E; }
    float alpha = ebuf[w] / (denom[d] + 1e-16f);
    const float4* hs = (const float4*)(h + (size_t)s * OUT_FT);
    float4 hv = hs[lane];
    float* od = out + (size_t)d * OUT_FT + lane * 4;
    atomicAdd(od + 0, alpha * hv.x);
    atomicAdd(od + 1, alpha * hv.y);
    atomicAdd(od + 2, alpha * hv.z);
    atomicAdd(od + 3, alpha * hv.w);
}

// ---------------------------------------------------------------------------
// Finalize: out = PReLU(out + gat_bias + bias)
// ---------------------------------------------------------------------------
__global__ void gat_final_kernel(float* __restrict__ out,
                                 const float* __restrict__ gat_bias,
                                 const float* __restrict__ bias,
                                 const float* __restrict__ prelu_a, int N) {
    int i = blockIdx.x * blockDim.x + threadIdx.x;
    if (i >= N * OUT_FT) return;
    int f = i & (OUT_FT - 1);
    float v = out[i] + gat_bias[f] + bias[f];
    float a = prelu_a[0];
    out[i] = (v >= 0.0f) ? v : a * v;
}

// ---------------------------------------------------------------------------
extern "C" void kernel_launch(void* const* d_in, const int* in_sizes, int n_in,
                              void* d_out, int out_size, void* d_ws, size_t ws_size,
                              hipStream_t stream) {
    const float*     seq      = (const float*)d_in[0];
    const long long* ei       = (const long long*)d_in[1];
    const float*     Wfc      = (const float*)d_in[2];
    const float*     Wgat     = (const float*)d_in[3];
    const float*     att_src  = (const float*)d_in[4];
    const float*     att_dst  = (const float*)d_in[5];
    const float*     gat_bias = (const float*)d_in[6];
    const float*     bias     = (const float*)d_in[7];
    const float*     prelu_a  = (const float*)d_in[8];
    float*           out      = (float*)d_out;

    const int N  = in_sizes[0] / IN_FT;      // 10000
    const int E  = in_sizes[1] / 2;          // 640000
    const int EP = E + N;                    // with self loops

    // workspace layout (floats)
    float*    h     = (float*)d_ws;                       // N*128
    float*    a_s   = h + (size_t)N * OUT_FT;             // N
    float*    a_d   = a_s + N;                            // N
    float*    denom = a_d + N;                            // N
    unsigned* maxb  = (unsigned*)(denom + N);             // N
    float*    ebuf  = (float*)(maxb + N);                 // EP

    const int TPB = 256;

    gat_init_kernel<<<(N * OUT_FT + TPB - 1) / TPB, TPB, 0, stream>>>(out, maxb, denom, N);

    // 8 waves/block, each wave a 16x16 WMMA tile; block covers 16 nodes x 128 feats
    gat_gemm_fused_kernel<<<N / 16, 256, 0, stream>>>(seq, Wfc, Wgat, h, N);

    gat_att_kernel<<<(N + 7) / 8, 256, 0, stream>>>(h, att_src, att_dst, a_s, a_d, N);

    gat_edge_max_kernel<<<(EP + TPB - 1) / TPB, TPB, 0, stream>>>(ei, a_s, a_d, ebuf, maxb, E, EP);

    gat_edge_exp_kernel<<<(EP + TPB - 1) / TPB, TPB, 0, stream>>>(ei, ebuf, maxb, denom, E, EP);

    gat_edge_scatter_kernel<<<(EP + 7) / 8, 256, 0, stream>>>(ei, ebuf, denom, h, out, E, EP);

    gat_final_kernel<<<(N * OUT_FT + TPB - 1) / TPB, TPB, 0, stream>>>(out, gat_bias, bias, prelu_a, N);
}